// GraphConvLayer_35107062678349
// MI455X (gfx1250) — compile-verified
//
#include <hip/hip_runtime.h>
#include <hip/hip_bf16.h>

typedef __attribute__((ext_vector_type(2))) float v2f;
typedef __attribute__((ext_vector_type(8))) float v8f;

#define N_NODES 100000
#define N_EDGES 3200000
#define D_FEAT  128

// ---------------------------------------------------------------------------
// Kernel 1: zero the accumulator workspace (sum[100000x128] ++ deg[100000]).
// ---------------------------------------------------------------------------
__global__ void gc_zero_kernel(float* __restrict__ p, int n) {
    int i = blockIdx.x * blockDim.x + threadIdx.x;
    if (i < n) p[i] = 0.0f;
}

// ---------------------------------------------------------------------------
// Kernel 2: edge scatter. One wave32 per edge; each lane moves a float4 of
// the source row and atomically accumulates into sum[dst]. Lane 0 bumps the
// degree counter. src/dst loads are wave-uniform (broadcast from one request).
// features + sum both fit in the 192MB L2, so this is L2-atomic bound.
// ---------------------------------------------------------------------------
__global__ void gc_scatter_kernel(const float* __restrict__ feat,
                                  const int* __restrict__ src,
                                  const int* __restrict__ dst,
                                  float* __restrict__ sum,
                                  float* __restrict__ deg,
                                  int nEdges) {
    const int lane = threadIdx.x & 31;
    const int edge = blockIdx.x * (blockDim.x >> 5) + (threadIdx.x >> 5);
    if (edge >= nEdges) return;

    const int s = src[edge];
    const int d = dst[edge];

    const float4 v = ((const float4*)(feat + (size_t)s * D_FEAT))[lane];
    float* o = sum + (size_t)d * D_FEAT + lane * 4;
    atomicAdd(o + 0, v.x);   // global_atomic_add_f32, non-returning
    atomicAdd(o + 1, v.y);
    atomicAdd(o + 2, v.z);
    atomicAdd(o + 3, v.w);
    if (lane == 0) atomicAdd(deg + d, 1.0f);
}

// ---------------------------------------------------------------------------
// Kernel 3: fused normalize + GEMM + bias + ReLU using V_WMMA_F32_16X16X4_F32.
// Block = 256 threads = 8 waves. Block b owns output rows [16b, 16b+16);
// wave w owns output cols [16w, 16w+16). The 16x128 H tile (mean-or-feature
// select) is built once in LDS and shared by all 8 waves; W (64KB) and bias
// are also staged in LDS.
// ---------------------------------------------------------------------------
__global__ void gc_gemm_kernel(const float* __restrict__ feat,
                               const float* __restrict__ W,
                               const float* __restrict__ bias,
                               const float* __restrict__ sum,
                               const float* __restrict__ deg,
                               float* __restrict__ out) {
    __shared__ float Wlds[D_FEAT * D_FEAT];   // 64 KB
    __shared__ float Hlds[16 * D_FEAT];       //  8 KB
    __shared__ float Blds[D_FEAT];            // 512 B

    const int tid = threadIdx.x;

    // Stage W into LDS: 4096 float4 across 256 threads.
    {
        const float4* w4  = (const float4*)W;
        float4*       wl4 = (float4*)Wlds;
        #pragma unroll
        for (int i = 0; i < 16; ++i)
            wl4[tid + i * 256] = w4[tid + i * 256];
        if (tid < D_FEAT) Blds[tid] = bias[tid];
    }

    // Stage H tile: h = deg>0 ? sum/deg : features. 512 float4 over 256 thr.
    const int rowBase = blockIdx.x * 16;
    {
        float4* hl4 = (float4*)Hlds;
        #pragma unroll
        for (int t = 0; t < 2; ++t) {
            const int i   = tid + t * 256;        // float4 index within tile
            const int r   = i >> 5;               // row in tile (128/4 = 32 f4 per row)
            const int c   = i & 31;               // float4 column
            const int row = rowBase + r;
            const float dg = deg[row];
            float4 v;
            if (dg > 0.0f) {
                const float inv = 1.0f / dg;
                float4 sv = ((const float4*)(sum + (size_t)row * D_FEAT))[c];
                v = make_float4(sv.x * inv, sv.y * inv, sv.z * inv, sv.w * inv);
            } else {
                v = ((const float4*)(feat + (size_t)row * D_FEAT))[c];
            }
            hl4[i] = v;
        }
    }
    __syncthreads();

    // WMMA: 16x16 tile per wave, K=128 in steps of 4.
    const int wave = tid >> 5;
    const int lane = tid & 31;
    const int half = lane >> 4;      // 0: lanes 0-15, 1: lanes 16-31
    const int l16  = lane & 15;
    const int col0 = wave * 16;

    v8f acc = {};
    #pragma unroll 4
    for (int k0 = 0; k0 < D_FEAT; k0 += 4) {
        const int ak = k0 + 2 * half;
        // A fragment (16x4): row = l16, K = {ak, ak+1}  -> ds_load_b64
        v2f a = *(const v2f*)(Hlds + l16 * D_FEAT + ak);
        // B fragment (4x16): rows {ak, ak+1}, col = col0 + l16
        v2f b;
        b.x = Wlds[ak * D_FEAT + col0 + l16];
        b.y = Wlds[(ak + 1) * D_FEAT + col0 + l16];
        acc = __builtin_amdgcn_wmma_f32_16x16x4_f32(
            /*neg_a=*/false, a, /*neg_b=*/false, b,
            /*c_mod=*/(short)0, acc, /*reuse_a=*/false, /*reuse_b=*/false);
    }

    // D layout: VGPR j -> row rowBase + j + 8*half, col = col0 + l16.
    const int col = col0 + l16;
    const float bb = Blds[col];
    const int row0 = rowBase + 8 * half;
    #pragma unroll
    for (int j = 0; j < 8; ++j) {
        float v = acc[j] + bb;
        out[(size_t)(row0 + j) * D_FEAT + col] = v > 0.0f ? v : 0.0f;
    }
}

// ---------------------------------------------------------------------------
// Launch: inputs = {features, W, b, src, dst}; out = relu(h @ W + b) f32.
// Workspace: sum (100000*128 f32) then deg (100000 f32) = ~51.6 MB.
// ---------------------------------------------------------------------------
extern "C" void kernel_launch(void* const* d_in, const int* in_sizes, int n_in,
                              void* d_out, int out_size, void* d_ws, size_t ws_size,
                              hipStream_t stream) {
    const float* feat = (const float*)d_in[0];
    const float* W    = (const float*)d_in[1];
    const float* bias = (const float*)d_in[2];
    const int*   src  = (const int*)d_in[3];
    const int*   dst  = (const int*)d_in[4];
    float*       out  = (float*)d_out;

    float* sum = (float*)d_ws;
    float* deg = sum + (size_t)N_NODES * D_FEAT;

    const int nZero = N_NODES * D_FEAT + N_NODES;
    gc_zero_kernel<<<(nZero + 255) / 256, 256, 0, stream>>>(sum, nZero);

    const int edgesPerBlock = 256 / 32;   // 8 waves -> 8 edges per block
    gc_scatter_kernel<<<(N_EDGES + edgesPerBlock - 1) / edgesPerBlock, 256, 0, stream>>>(
        feat, src, dst, sum, deg, N_EDGES);

    gc_gemm_kernel<<<N_NODES / 16, 256, 0, stream>>>(feat, W, bias, sum, deg, out);
}